// DiscriminativeLoss_52647709114533
// MI455X (gfx1250) — compile-verified
//
#include <hip/hip_runtime.h>
#include <math.h>

// Problem constants (match reference)
#define BATCH    8
#define EDIM     32
#define HW       (512 * 512)
#define NBIN     33            // instances 0..32 (0 = background)
#define STRIDEP  33            // LDS padding stride to spread banks
#define PER_SAMPLE_VALS (NBIN * EDIM)   // 1056

// Pass tiling: 256 threads * 4 px (float4) * 8 iters = 8192 px/block -> 32 blocks/sample
#define THREADS  256
#define PX_PER_THREAD_VEC 4
#define ITERS    8
#define BLOCKS_PER_SAMPLE (HW / (THREADS * PX_PER_THREAD_VEC * ITERS))  // 32

typedef __attribute__((ext_vector_type(2))) float v2f;
typedef __attribute__((ext_vector_type(8))) float v8f;

// ---------------------------------------------------------------- zero ws
__global__ void k_zero(float* __restrict__ ws, int n) {
  int i = blockIdx.x * blockDim.x + threadIdx.x;
  if (i < n) ws[i] = 0.0f;
}

// ------------------------------------------------- pass 1: segment sums/counts
__global__ void k_pass1(const float* __restrict__ emb, const int* __restrict__ masks,
                        float* __restrict__ g_sums, float* __restrict__ g_counts) {
  __shared__ float lsum[NBIN * STRIDEP];  // [inst*33 + e], padded vs. 64 banks
  __shared__ float lcnt[NBIN];

  const int tid = threadIdx.x;
  const int b   = blockIdx.y;
  const int base = blockIdx.x * (THREADS * PX_PER_THREAD_VEC * ITERS);

  for (int i = tid; i < NBIN * STRIDEP; i += THREADS) lsum[i] = 0.0f;
  if (tid < NBIN) lcnt[tid] = 0.0f;
  __syncthreads();

  const float* embB  = emb   + (size_t)b * EDIM * HW;
  const int*   maskB = masks + (size_t)b * HW;

  for (int it = 0; it < ITERS; ++it) {
    const int p = base + it * (THREADS * PX_PER_THREAD_VEC) + tid * PX_PER_THREAD_VEC;
    const int4 m4 = *(const int4*)(maskB + p);
#pragma unroll
    for (int e = 0; e < EDIM; ++e) {
      const float4 v = *(const float4*)(embB + (size_t)e * HW + p);
      atomicAdd(&lsum[m4.x * STRIDEP + e], v.x);
      atomicAdd(&lsum[m4.y * STRIDEP + e], v.y);
      atomicAdd(&lsum[m4.z * STRIDEP + e], v.z);
      atomicAdd(&lsum[m4.w * STRIDEP + e], v.w);
    }
    atomicAdd(&lcnt[m4.x], 1.0f);
    atomicAdd(&lcnt[m4.y], 1.0f);
    atomicAdd(&lcnt[m4.z], 1.0f);
    atomicAdd(&lcnt[m4.w], 1.0f);
  }
  __syncthreads();

  for (int idx = tid; idx < PER_SAMPLE_VALS; idx += THREADS) {
    const int inst = idx >> 5, e = idx & 31;
    atomicAdd(&g_sums[b * PER_SAMPLE_VALS + idx], lsum[inst * STRIDEP + e]);
  }
  if (tid < NBIN) atomicAdd(&g_counts[b * NBIN + tid], lcnt[tid]);
}

// ------------------- means + Gram (WMMA f32 16x16x4) -> dist_loss, reg_loss
__global__ void k_means_dist_reg(const float* __restrict__ g_sums,
                                 const float* __restrict__ g_counts,
                                 float* __restrict__ g_means,
                                 float* __restrict__ g_distl,
                                 float* __restrict__ g_regl) {
  const int b    = blockIdx.x;      // one sample per block
  const int lane = threadIdx.x;     // 32 threads == one wave32, EXEC all-1s

  __shared__ float mu[32][STRIDEP]; // instance means 1..32 (row = inst-1)
  __shared__ float r2[32];          // squared norms
  __shared__ float red[2];          // {dist_sum, reg_sum}

  if (lane < 2) red[lane] = 0.0f;

  // means_all = sums / max(counts,1); publish to global, stage fg means in LDS
  for (int idx = lane; idx < PER_SAMPLE_VALS; idx += 32) {
    const int inst = idx >> 5, e = idx & 31;
    const float s = g_sums[b * PER_SAMPLE_VALS + idx];
    const float c = g_counts[b * NBIN + inst];
    const float m = s / fmaxf(c, 1.0f);
    g_means[b * PER_SAMPLE_VALS + idx] = m;
    if (inst >= 1) mu[inst - 1][e] = m;
  }
  __syncthreads();

  float acc = 0.0f;
#pragma unroll
  for (int k = 0; k < 32; ++k) { const float v = mu[lane][k]; acc += v * v; }
  r2[lane] = acc;
  __syncthreads();

  // Gram = Mu * Mu^T via V_WMMA_F32_16X16X4_F32.
  // A 16x4 f32 layout: lanes 0-15 row M=lane (K=k,k+1), lanes 16-31 (K=k+2,k+3).
  // B 4x16 mirrors it; since B = Mu^T, both fragments read mu[] identically.
  const int l15 = lane & 15;
  const int hi  = lane >> 4;
  float dsum = 0.0f;

  for (int mt = 0; mt < 2; ++mt) {
    for (int nt = 0; nt < 2; ++nt) {
      v8f c = {};
#pragma unroll
      for (int k = 0; k < 32; k += 4) {
        v2f a, bb;
        a.x  = mu[mt * 16 + l15][k + 2 * hi];
        a.y  = mu[mt * 16 + l15][k + 2 * hi + 1];
        bb.x = mu[nt * 16 + l15][k + 2 * hi];
        bb.y = mu[nt * 16 + l15][k + 2 * hi + 1];
        c = __builtin_amdgcn_wmma_f32_16x16x4_f32(
            /*neg_a=*/false, a, /*neg_b=*/false, bb,
            /*c_mod=*/(short)0, c, /*reuse_a=*/false, /*reuse_b=*/false);
      }
      // C/D layout: VGPR r, lanes 0-15 -> M = r, lanes 16-31 -> M = 8 + r; N = l15
#pragma unroll
      for (int r = 0; r < 8; ++r) {
        const int i = mt * 16 + hi * 8 + r;
        const int j = nt * 16 + l15;
        if (j > i) {  // strict upper triangle
          const float sq = r2[i] + r2[j] - 2.0f * c[r];
          const float pd = sqrtf(fmaxf(sq, 1e-12f));
          const float h  = fmaxf(3.0f - pd, 0.0f);  // 2*DELTA_D = 3.0
          dsum += h * h;
        }
      }
    }
  }
  atomicAdd(&red[0], dsum);
  atomicAdd(&red[1], sqrtf(r2[lane] + 1e-12f));
  __syncthreads();
  if (lane == 0) {
    g_distl[b] = red[0] * (1.0f / 496.0f);  // 32*31/2 pairs
    g_regl[b]  = red[1] * (1.0f / 32.0f);
  }
}

// --------------------------------------- pass 2: per-pixel variance hinge
__global__ void k_pass2(const float* __restrict__ emb, const int* __restrict__ masks,
                        const float* __restrict__ g_means, float* __restrict__ g_hinge) {
  __shared__ float lmu[NBIN * STRIDEP];
  __shared__ float lh[NBIN];

  const int tid = threadIdx.x;
  const int b   = blockIdx.y;
  const int base = blockIdx.x * (THREADS * PX_PER_THREAD_VEC * ITERS);

  for (int idx = tid; idx < PER_SAMPLE_VALS; idx += THREADS) {
    const int inst = idx >> 5, e = idx & 31;
    lmu[inst * STRIDEP + e] = g_means[b * PER_SAMPLE_VALS + idx];
  }
  if (tid < NBIN) lh[tid] = 0.0f;
  __syncthreads();

  const float* embB  = emb   + (size_t)b * EDIM * HW;
  const int*   maskB = masks + (size_t)b * HW;

  for (int it = 0; it < ITERS; ++it) {
    const int p = base + it * (THREADS * PX_PER_THREAD_VEC) + tid * PX_PER_THREAD_VEC;
    const int4 m4 = *(const int4*)(maskB + p);
    float d0 = 0.0f, d1 = 0.0f, d2 = 0.0f, d3 = 0.0f;
#pragma unroll
    for (int e = 0; e < EDIM; ++e) {
      const float4 v = *(const float4*)(embB + (size_t)e * HW + p);
      float t;
      t = v.x - lmu[m4.x * STRIDEP + e]; d0 += t * t;
      t = v.y - lmu[m4.y * STRIDEP + e]; d1 += t * t;
      t = v.z - lmu[m4.z * STRIDEP + e]; d2 += t * t;
      t = v.w - lmu[m4.w * STRIDEP + e]; d3 += t * t;
    }
    float h;
    if (m4.x > 0) { h = fmaxf(sqrtf(d0 + 1e-12f) - 0.5f, 0.0f); atomicAdd(&lh[m4.x], h * h); }
    if (m4.y > 0) { h = fmaxf(sqrtf(d1 + 1e-12f) - 0.5f, 0.0f); atomicAdd(&lh[m4.y], h * h); }
    if (m4.z > 0) { h = fmaxf(sqrtf(d2 + 1e-12f) - 0.5f, 0.0f); atomicAdd(&lh[m4.z], h * h); }
    if (m4.w > 0) { h = fmaxf(sqrtf(d3 + 1e-12f) - 0.5f, 0.0f); atomicAdd(&lh[m4.w], h * h); }
  }
  __syncthreads();
  if (tid >= 1 && tid < NBIN) atomicAdd(&g_hinge[b * NBIN + tid], lh[tid]);
}

// ------------------------------------------------------------- final combine
__global__ void k_final(const float* __restrict__ g_counts, const float* __restrict__ g_hinge,
                        const float* __restrict__ g_distl, const float* __restrict__ g_regl,
                        float* __restrict__ out) {
  __shared__ float npres[BATCH], vsum[BATCH], tot[3];
  const int tid = threadIdx.x;
  if (tid < BATCH) { npres[tid] = 0.0f; vsum[tid] = 0.0f; }
  if (tid < 3) tot[tid] = 0.0f;
  __syncthreads();

  // 8 samples * 32 instances = 256 == blockDim
  {
    const int b = tid >> 5;
    const int i = (tid & 31) + 1;
    const float c = g_counts[b * NBIN + i];
    if (c > 0.0f) {
      atomicAdd(&npres[b], 1.0f);
      atomicAdd(&vsum[b], g_hinge[b * NBIN + i] / c);
    }
  }
  __syncthreads();
  if (tid < BATCH) {
    atomicAdd(&tot[0], vsum[tid] / fmaxf(npres[tid], 1.0f));
    atomicAdd(&tot[1], g_distl[tid]);
    atomicAdd(&tot[2], g_regl[tid]);
  }
  __syncthreads();
  if (tid == 0) {
    const float var_l  = tot[0] * (1.0f / BATCH);
    const float dist_l = tot[1] * (1.0f / BATCH);
    const float reg_l  = tot[2] * (1.0f / BATCH);
    out[0] = var_l + dist_l + 0.001f * reg_l;  // ALPHA=BETA=1, GAMMA=1e-3
    out[1] = var_l;
    out[2] = dist_l;
    out[3] = reg_l;
  }
}

// ---------------------------------------------------------------- launcher
extern "C" void kernel_launch(void* const* d_in, const int* in_sizes, int n_in,
                              void* d_out, int out_size, void* d_ws, size_t ws_size,
                              hipStream_t stream) {
  const float* emb   = (const float*)d_in[0];   // (8,32,512,512) f32
  const int*   masks = (const int*)d_in[1];     // (8,512,512) i32
  float* out = (float*)d_out;                   // 4 floats
  float* ws  = (float*)d_ws;

  // workspace layout (floats)
  float* g_sums   = ws;                                    // 8*1056
  float* g_counts = g_sums   + BATCH * PER_SAMPLE_VALS;    // 8*33
  float* g_means  = g_counts + BATCH * NBIN;               // 8*1056
  float* g_hinge  = g_means  + BATCH * PER_SAMPLE_VALS;    // 8*33
  float* g_distl  = g_hinge  + BATCH * NBIN;               // 8
  float* g_regl   = g_distl  + BATCH;                      // 8
  const int nzero = (int)((g_regl + BATCH) - ws);

  k_zero<<<(nzero + THREADS - 1) / THREADS, THREADS, 0, stream>>>(ws, nzero);

  dim3 grid(BLOCKS_PER_SAMPLE, BATCH);
  k_pass1<<<grid, THREADS, 0, stream>>>(emb, masks, g_sums, g_counts);
  k_means_dist_reg<<<BATCH, 32, 0, stream>>>(g_sums, g_counts, g_means, g_distl, g_regl);
  k_pass2<<<grid, THREADS, 0, stream>>>(emb, masks, g_means, g_hinge);
  k_final<<<1, THREADS, 0, stream>>>(g_counts, g_hinge, g_distl, g_regl, out);
}